// Stack_84782654423689
// MI455X (gfx1250) — compile-verified
//
#include <hip/hip_runtime.h>
#include <math.h>

// dims from the reference
#define BB 4
#define HH 8
#define PP 32
#define NN 32
#define DD 16
#define LL 1024          // PP*NN
#define EE 2
#define WINW 10
#define NFF 6
#define PREDN 96
#define SEQN 512
#define ST 17            // LDS row stride in floats (16 + 1 pad, conflict-free)
#define EPSF 1.1920928955078125e-07f  // float32 finfo eps

typedef __attribute__((ext_vector_type(2))) float v2f;
typedef __attribute__((ext_vector_type(8))) float v8f;

// V_WMMA_F32_16X16X4_F32: D = A(16x4 f32) * B(4x16 f32) + C(16x16 f32)
static __device__ __forceinline__ v8f wmma4(v2f a, v2f b, v8f c) {
  return __builtin_amdgcn_wmma_f32_16x16x4_f32(false, a, false, b, (short)0, c,
                                               false, false);
}

// DPP16 lane permute (EXEC is all-ones wherever these are used)
template <int CTRL>
static __device__ __forceinline__ float dppf(float v) {
  const int r = __builtin_amdgcn_mov_dpp(__builtin_bit_cast(int, v), CTRL,
                                         0xF, 0xF, true);
  return __builtin_bit_cast(float, r);
}
// butterfly reduction across the 16 lanes of a wave32 half:
// quad_perm{1,0,3,2}=0xB1 (xor1), quad_perm{2,3,0,1}=0x4E (xor2),
// row_half_mirror=0x141 (pairs 4-groups in 8), row_mirror=0x140 (pairs 8s in 16)
static __device__ __forceinline__ float redmax16(float v) {
  v = fmaxf(v, dppf<0xB1>(v));
  v = fmaxf(v, dppf<0x4E>(v));
  v = fmaxf(v, dppf<0x141>(v));
  v = fmaxf(v, dppf<0x140>(v));
  return v;
}
static __device__ __forceinline__ float redmin16(float v) {
  v = fminf(v, dppf<0xB1>(v));
  v = fminf(v, dppf<0x4E>(v));
  v = fminf(v, dppf<0x141>(v));
  v = fminf(v, dppf<0x140>(v));
  return v;
}
static __device__ __forceinline__ float redsum16(float v) {
  v += dppf<0xB1>(v);
  v += dppf<0x4E>(v);
  v += dppf<0x141>(v);
  v += dppf<0x140>(v);
  return v;
}
static __device__ __forceinline__ float gelu_exact(float x) {
  return 0.5f * x * (1.0f + erff(x * 0.70710678118654752440f));
}

// One workgroup per (b,h) stream; entire L x D state lives in LDS for both
// layers. buf0 -> attention -> buf1 -> convFFN -> buf0 (ping-pong).
__global__ __launch_bounds__(256) void ts_layers_kernel(
    const float* __restrict__ tokens,        // [B,H,L,D]
    const float* __restrict__ log_scales,    // [E]
    const float* __restrict__ attn_norm_w,   // [E,D]
    const float* __restrict__ conv_up_w,     // [E,2H,1,3]
    const float* __restrict__ conv_down_w,   // [E,H,2,1]
    const float* __restrict__ ffn_norm_w,    // [E,D]
    float* __restrict__ hws)                 // [B,H,L,D] scratch out
{
  __shared__ float buf0[LL * ST];
  __shared__ float buf1[LL * ST];
  __shared__ float pscr[8 * 16 * ST];  // per-wave P(+) tile scratch
  __shared__ float qscr[8 * 16 * ST];  // per-wave P(-) tile scratch

  const int tid = threadIdx.x;
  const int lid = tid & 31;
  // wave id is uniform: force it into an SGPR so band loops/branches are
  // scalar and EXEC is structurally all-ones around every WMMA.
  const int wv  = __builtin_amdgcn_readfirstlane(tid >> 5);  // 0..7
  const int ln  = lid & 15;   // lane within half
  const int hf  = lid >> 4;   // half 0/1
  const int bh  = blockIdx.x; // b*HH + head
  const int head = bh & (HH - 1);

  // tokens -> buf0
  const float* tsrc = tokens + (size_t)bh * LL * DD;
  for (int g = tid; g < LL * DD; g += 256)
    buf0[(g >> 4) * ST + (g & 15)] = tsrc[g];
  __syncthreads();

  float* ps = pscr + wv * 16 * ST;
  float* qs = qscr + wv * 16 * ST;

  for (int layer = 0; layer < EE; ++layer) {
    // ================= signed banded attention + residual + RMSNorm =======
    const float scale =
        fminf(fmaxf(__expf(log_scales[layer]), 1.0f), 30.0f) * 0.25f; // /sqrt(16)
    const float wa = attn_norm_w[layer * DD + ln];

    for (int t = wv; t < LL / 16; t += 8) {       // 16-row query tiles
      const int q0  = t * 16;
      const int pid = t >> 1;                     // patch of this tile
      const int kb  = (pid + 1) * NN;             // band: pid+1 .. pid+9
      const int ke  = min(pid + WINW, PP) * NN;

      // Q A-fragments (ISA fp32 A layout: lane=M, VGPR0=K0/K2, VGPR1=K1/K3)
      v2f aq[4];
#pragma unroll
      for (int j = 0; j < 4; ++j) {
        const int col = 4 * j + 2 * hf;
        aq[j][0] = buf0[(q0 + ln) * ST + col];
        aq[j][1] = buf0[(q0 + ln) * ST + col + 1];
      }

      v8f outt = {0, 0, 0, 0, 0, 0, 0, 0};
      if (kb < ke) {
        // ---- pass A: exact per-row max/min of raw scores over the band ----
        float rmx[8], rmn[8];
#pragma unroll
        for (int r = 0; r < 8; ++r) { rmx[r] = -3.0e38f; rmn[r] = 3.0e38f; }
        for (int k0 = kb; k0 < ke; k0 += 16) {
          v2f bk[4];  // B = K^T: b[k,n] = K[n,k] -> same access pattern as A
#pragma unroll
          for (int j = 0; j < 4; ++j) {
            const int col = 4 * j + 2 * hf;
            bk[j][0] = buf0[(k0 + ln) * ST + col];
            bk[j][1] = buf0[(k0 + ln) * ST + col + 1];
          }
          v8f c = {0, 0, 0, 0, 0, 0, 0, 0};
#pragma unroll
          for (int j = 0; j < 4; ++j) c = wmma4(aq[j], bk[j], c);
#pragma unroll
          for (int r = 0; r < 8; ++r) {
            rmx[r] = fmaxf(rmx[r], c[r]);
            rmn[r] = fminf(rmn[r], c[r]);
          }
        }
        float Mp[8], Mn[8];
#pragma unroll
        for (int r = 0; r < 8; ++r) {
          Mp[r] = scale * redmax16(rmx[r]);   // max of +scale*s
          Mn[r] = -scale * redmin16(rmn[r]);  // max of -scale*s
        }

        // ---- pass B: exp, row sums, and P(+/-) @ V via WMMA ----
        v8f opos = {0, 0, 0, 0, 0, 0, 0, 0};
        v8f oneg = {0, 0, 0, 0, 0, 0, 0, 0};
        float zp[8], zn[8];
#pragma unroll
        for (int r = 0; r < 8; ++r) { zp[r] = 0.0f; zn[r] = 0.0f; }

        for (int k0 = kb; k0 < ke; k0 += 16) {
          v2f bk[4];
#pragma unroll
          for (int j = 0; j < 4; ++j) {
            const int col = 4 * j + 2 * hf;
            bk[j][0] = buf0[(k0 + ln) * ST + col];
            bk[j][1] = buf0[(k0 + ln) * ST + col + 1];
          }
          v8f c = {0, 0, 0, 0, 0, 0, 0, 0};
#pragma unroll
          for (int j = 0; j < 4; ++j) c = wmma4(aq[j], bk[j], c);

          // exponentials in C layout; stage tiles in LDS for A-layout reload
#pragma unroll
          for (int r = 0; r < 8; ++r) {
            const float sv = scale * c[r];
            const float pe = __expf(sv - Mp[r]);
            const float qe = __expf(-sv - Mn[r]);
            zp[r] += pe;
            zn[r] += qe;
            ps[(r + 8 * hf) * ST + ln] = pe;
            qs[(r + 8 * hf) * ST + ln] = qe;
          }
          asm volatile("s_wait_dscnt 0" ::: "memory");  // cross-lane transpose

          v2f ap[4], an[4], bv[4];
#pragma unroll
          for (int j = 0; j < 4; ++j) {
            const int col = 4 * j + 2 * hf;
            ap[j][0] = ps[ln * ST + col];
            ap[j][1] = ps[ln * ST + col + 1];
            an[j][0] = qs[ln * ST + col];
            an[j][1] = qs[ln * ST + col + 1];
            // B = V chunk: b[k,n] = V[k0+k, d=n]
            bv[j][0] = buf0[(k0 + col) * ST + ln];
            bv[j][1] = buf0[(k0 + col + 1) * ST + ln];
          }
#pragma unroll
          for (int j = 0; j < 4; ++j) opos = wmma4(ap[j], bv[j], opos);
#pragma unroll
          for (int j = 0; j < 4; ++j) oneg = wmma4(an[j], bv[j], oneg);
        }
        // normalize: A = softmax(+) - softmax(-)
#pragma unroll
        for (int r = 0; r < 8; ++r) {
          const float Zp = redsum16(zp[r]);
          const float Zn = redsum16(zn[r]);
          outt[r] = opos[r] / Zp - oneg[r] / Zn;
        }
      }
      // residual + RMSNorm over D, write to buf1 (C layout: row = r + 8*hf)
#pragma unroll
      for (int r = 0; r < 8; ++r) {
        const int row = q0 + r + 8 * hf;
        const float x = buf0[row * ST + ln] + outt[r];
        const float ss = redsum16(x * x) * (1.0f / 16.0f);
        buf1[row * ST + ln] = x * rsqrtf(ss + EPSF) * wa;
      }
    }
    __syncthreads();

    // ======= depthwise causal conv FFN + residual + RMSNorm (buf1->buf0) ===
    const float* up = conv_up_w + (layer * 2 * HH + 2 * head) * 3;
    const float u00 = up[0], u01 = up[1], u02 = up[2];
    const float u10 = up[3], u11 = up[4], u12 = up[5];
    const float* dnw = conv_down_w + (layer * HH + head) * 2;
    const float dw0 = dnw[0], dw1 = dnw[1];

    for (int l = tid; l < LL; l += 256) {
      float x[DD], y[DD];
#pragma unroll
      for (int d = 0; d < DD; ++d) x[d] = buf1[l * ST + d];
      float ss = 0.0f;
#pragma unroll
      for (int d = 0; d < DD; ++d) {
        const float xm2 = (d >= 2) ? x[d - 2] : 0.0f;  // causal left pad 2
        const float xm1 = (d >= 1) ? x[d - 1] : 0.0f;
        const float g0 = gelu_exact(u00 * xm2 + u01 * xm1 + u02 * x[d]);
        const float g1 = gelu_exact(u10 * xm2 + u11 * xm1 + u12 * x[d]);
        y[d] = x[d] + dw0 * g0 + dw1 * g1;
        ss += y[d] * y[d];
      }
      const float inv = rsqrtf(ss * (1.0f / 16.0f) + EPSF);
#pragma unroll
      for (int d = 0; d < DD; ++d)
        buf0[l * ST + d] = y[d] * inv * ffn_norm_w[layer * DD + d];
    }
    __syncthreads();
  }

  // final hidden state -> global workspace
  float* hdst = hws + (size_t)bh * LL * DD;
  for (int g = tid; g < LL * DD; g += 256)
    hdst[g] = buf0[(g >> 4) * ST + (g & 15)];
}

// head mixing + forecast projection; forecast[b,t,n] with t = d*NF + fi
// (D*NF == PRED, so the reference tail-slice keeps everything)
__global__ __launch_bounds__(256) void mix_fore_kernel(
    const float* __restrict__ hws, const float* __restrict__ mix_w,
    const float* __restrict__ mix_b, const float* __restrict__ fore_w,
    const float* __restrict__ fore_b, float* __restrict__ out) {
  const int idx = blockIdx.x * blockDim.x + threadIdx.x;
  if (idx >= BB * NN * DD * NFF) return;
  const int fi = idx % NFF;
  const int d  = (idx / NFF) % DD;
  const int n  = (idx / (NFF * DD)) % NN;
  const int b  = idx / (NFF * DD * NN);

  float mw[HH];
#pragma unroll
  for (int h = 0; h < HH; ++h) mw[h] = mix_w[h];
  const float mb = mix_b[0];

  float acc = fore_b[fi];
  for (int p = 0; p < PP; ++p) {
    float m = mb;
#pragma unroll
    for (int h = 0; h < HH; ++h)
      m += hws[((size_t)(b * HH + h) * LL + p * NN + n) * DD + d] * mw[h];
    acc += m * fore_w[fi * PP + p];
  }
  out[(b * PREDN + (d * NFF + fi)) * NN + n] = acc;
}

extern "C" void kernel_launch(void* const* d_in, const int* in_sizes, int n_in,
                              void* d_out, int out_size, void* d_ws,
                              size_t ws_size, hipStream_t stream) {
  const float* tokens      = (const float*)d_in[0];
  const float* x_original  = (const float*)d_in[1];
  const float* log_scales  = (const float*)d_in[2];
  const float* attn_norm_w = (const float*)d_in[3];
  const float* conv_up_w   = (const float*)d_in[4];
  const float* conv_down_w = (const float*)d_in[5];
  const float* ffn_norm_w  = (const float*)d_in[6];
  const float* mix_w       = (const float*)d_in[7];
  const float* mix_b       = (const float*)d_in[8];
  const float* fore_w      = (const float*)d_in[9];
  const float* fore_b      = (const float*)d_in[10];
  float* out = (float*)d_out;
  float* hws = (float*)d_ws;  // needs B*H*L*D*4 = 2 MB

  ts_layers_kernel<<<dim3(BB * HH), dim3(256), 0, stream>>>(
      tokens, log_scales, attn_norm_w, conv_up_w, conv_down_w, ffn_norm_w, hws);

  mix_fore_kernel<<<dim3((BB * NN * DD * NFF + 255) / 256), dim3(256), 0,
                    stream>>>(hws, mix_w, mix_b, fore_w, fore_b, out);

  // tuple output: (forecast, x_original) concatenated flat
  hipMemcpyAsync(out + BB * PREDN * NN, x_original,
                 (size_t)BB * SEQN * NN * sizeof(float),
                 hipMemcpyDeviceToDevice, stream);
}